// ZeroPad2d_86191403696260
// MI455X (gfx1250) — compile-verified
//
#include <hip/hip_runtime.h>
#include <stdint.h>

// ZeroPad2d (pad=1): x (32, 2048, 2048) fp32 -> out (32, 2050, 2050) fp32.
// Pure bandwidth (~1.07 GB traffic -> ~46 us roofline at 23.3 TB/s).
// Interior copy: CDNA5 Tensor Data Mover, one D# pair per 8x2048 tile:
//   global->LDS (row stride 2048), s_wait_tensorcnt, LDS->global (row stride 2050,
//   dest offset +1 row +1 col). TDM handles the misaligned strided placement.
// Border zeros are written by the wave's lanes WHILE the tensor load is in flight
// (STOREcnt is independent of TENSORcnt), fusing everything into one launch.

#define NC 32
#define H  2048
#define W  2048
#define OH 2050
#define OW 2050
#define TR 8                 // tile rows: 8*2048*4 = 64 KB LDS -> 5 WGs/WGP concurrency

typedef uint32_t u32x4 __attribute__((ext_vector_type(4)));
typedef uint32_t u32x8 __attribute__((ext_vector_type(8)));

// ---- D# packing (CDNA5 ISA) ----
// Group 0 (128b): [1:0]=count=1, [63:32]=lds_addr, [120:64]=global_addr, [127:126]=type=2
__device__ __forceinline__ u32x4 tdm_group0(uint64_t gaddr, uint32_t lds_byte_off) {
  u32x4 g;
  g.x = 1u;                                   // count=1, is_restore=0, gather off
  g.y = lds_byte_off;                         // lds_addr (bytes)
  g.z = (uint32_t)gaddr;                      // global_addr[31:0]
  g.w = ((uint32_t)(gaddr >> 32) & 0x01FFFFFFu) | (2u << 30); // addr[56:32] | type=2
  return g;
}

// Group 1 (256b): data_size=4B, tensor dims, tile dims, dim0 stride (element units)
__device__ __forceinline__ u32x8 tdm_group1(uint32_t tdim0, uint32_t tdim1,
                                            uint32_t tile0, uint32_t tile1,
                                            uint64_t stride0) {
  u32x8 g;
  g.s0 = 2u << 16;                                  // workgroup_mask=0, data_size=2 (4B)
  g.s1 = (tdim0 & 0xFFFFu) << 16;                   // tensor_dim0[15:0] @ bits 63:48
  g.s2 = (tdim0 >> 16) | ((tdim1 & 0xFFFFu) << 16); // tensor_dim0[31:16], tensor_dim1[15:0]
  g.s3 = (tdim1 >> 16) | (tile0 << 16);             // tensor_dim1[31:16], tile_dim0
  g.s4 = tile1;                                     // tile_dim1 ; tile_dim2=0 (2D tile)
  g.s5 = (uint32_t)stride0;                         // tensor_dim0_stride[31:0]
  g.s6 = (uint32_t)(stride0 >> 32) & 0xFFFFu;       // stride0[47:32] ; stride1=0
  g.s7 = 0u;
  return g;
}

// One workgroup (1 wave) per 8x2048 tile: TDM moves the tile, lanes zero borders.
__global__ __launch_bounds__(32) void pad_tdm_fused(const float* __restrict__ in,
                                                    float* __restrict__ out) {
  __shared__ float tile[TR * W];                    // 64 KB staging
  // LDS byte offset of `tile` = low 32 bits of its generic address (only alloc -> 0).
  const uint32_t lds_off = (uint32_t)(uintptr_t)(void*)tile;

  const int tilesPerC = H / TR;                     // 256
  const int c    = blockIdx.x / tilesPerC;
  const int t    = blockIdx.x % tilesPerC;
  const int row0 = t * TR;

  const uint64_t src = (uint64_t)(in  + (size_t)c * H  * W  + (size_t)row0 * W);
  const uint64_t dst = (uint64_t)(out + (size_t)c * OH * OW + (size_t)(row0 + 1) * OW + 1);

  const u32x4 g0l = tdm_group0(src, lds_off);
  const u32x8 g1l = tdm_group1(W, TR, W, TR, (uint64_t)W);    // contiguous, stride 2048
  const u32x4 g0s = tdm_group0(dst, lds_off);
  const u32x8 g1s = tdm_group1(W, TR, W, TR, (uint64_t)OW);   // strided, stride 2050
  const u32x4 gz  = {0u, 0u, 0u, 0u};               // groups 2/3: dims 3/4 unused

  // Kick off Global -> LDS DMA (TENSORcnt++); wave continues immediately.
  asm volatile("tensor_load_to_lds %0, %1, %2, %3"
               :: "s"(g0l), "s"(g1l), "s"(gz), "s"(gz)
               : "memory");

  // ---- Border zeros, overlapped with the DMA (tracked by STOREcnt, not TENSORcnt) ----
  const int lane = threadIdx.x;
  float* obase = out + (size_t)c * OH * OW;
  // Left/right columns for this tile's TR output rows (rows row0+1 .. row0+TR).
  if (lane < 2 * TR) {
    const int r = lane >> 1;
    obase[(size_t)(row0 + 1 + r) * OW + ((lane & 1) ? (OW - 1) : 0)] = 0.0f;
  }
  // First tile of a channel also zeroes the full top row (y = 0, incl. corners).
  if (t == 0) {
    for (int i = lane; i < OW; i += 32) obase[i] = 0.0f;
  }
  // Last tile of a channel zeroes the full bottom row (y = OH-1).
  if (t == tilesPerC - 1) {
    float* brow = obase + (size_t)(OH - 1) * OW;
    for (int i = lane; i < OW; i += 32) brow[i] = 0.0f;
  }

  // Wait for the tile to land in LDS, then LDS -> Global with output stride 2050.
  __builtin_amdgcn_s_wait_tensorcnt(0);
  asm volatile("tensor_store_from_lds %0, %1, %2, %3"
               :: "s"(g0s), "s"(g1s), "s"(gz), "s"(gz)
               : "memory");
  // S_ENDPGM implies wait-idle; explicit drain is free and self-documenting.
  __builtin_amdgcn_s_wait_tensorcnt(0);
}

extern "C" void kernel_launch(void* const* d_in, const int* in_sizes, int n_in,
                              void* d_out, int out_size, void* d_ws, size_t ws_size,
                              hipStream_t stream) {
  (void)in_sizes; (void)n_in; (void)out_size; (void)d_ws; (void)ws_size;
  const float* x = (const float*)d_in[0];
  float* out     = (float*)d_out;

  pad_tdm_fused<<<dim3(NC * (H / TR)), dim3(32), 0, stream>>>(x, out);
}